// Generator_1228360646784
// MI455X (gfx1250) — compile-verified
//
#include <hip/hip_runtime.h>

// CDNA5 / gfx1250 fused structural-MLP kernel.
// D = A(16x4 f32) x B(4x16 f32) + C(16x16 f32) via V_WMMA_F32_16X16X4_F32.
typedef __attribute__((ext_vector_type(2))) float v2f;
typedef __attribute__((ext_vector_type(8))) float v8f;

#define STATE 6
#define HID 64
#define PAD 66              // padded row stride (floats) -> conflict-free LDS
#define WAVES 4
#define ROWS_PER_WAVE 16
#define ROWS_PER_BLOCK (WAVES * ROWS_PER_WAVE)
#define BLOCK 128
#define SLOPE 0.2f

// ---- LDS layout (float offsets, all even => 8B aligned for b64 ds ops) ----
#define OFF_W1T 0
#define SZ_W1T  (STATE * HID * PAD)            // 25344 : W1 transposed [s][n][k]+pad
#define OFF_W0  (OFF_W1T + SZ_W1T)             // 25344
#define SZ_W0   (STATE * 4 * HID)              // 1536  : W0 [s][k=4][n=64]
#define OFF_B0  (OFF_W0 + SZ_W0)               // 26880
#define OFF_B1  (OFF_B0 + STATE * HID)         // 27264
#define OFF_W2  (OFF_B1 + STATE * HID)         // 27648
#define OFF_B2  (OFF_W2 + STATE * HID)         // 28032 (6 used, pad to 8)
#define OFF_PIDX (OFF_B2 + 8)                  // 28040 (18 ints, pad to 20)
#define OFF_YZ  (OFF_PIDX + 20)                // 28060 : per wave 96 y + 96 z
#define OFF_H0  (OFF_YZ + WAVES * 192)         // 28828 : per wave 16 x PAD
#define SMEM_FLOATS (OFF_H0 + WAVES * ROWS_PER_WAVE * PAD)  // 33052 -> ~129 KB

__device__ __forceinline__ float leaky(float v) {
    // SLOPE < 1  =>  leaky(v) == max(v, SLOPE*v)   (one mul + one max, packs to pk ops)
    return fmaxf(v, SLOPE * v);
}

__global__ void __launch_bounds__(BLOCK)
scm_mlp_wmma_kernel(const float* __restrict__ y, const float* __restrict__ z,
                    const int* __restrict__ pidx_g,
                    const float* __restrict__ W0, const float* __restrict__ b0,
                    const float* __restrict__ W1, const float* __restrict__ b1,
                    const float* __restrict__ W2, const float* __restrict__ b2,
                    float* __restrict__ out, int Bn)
{
    extern __shared__ float smem[];
    float* sW1T = smem + OFF_W1T;
    float* sW0  = smem + OFF_W0;
    float* sB0  = smem + OFF_B0;
    float* sB1  = smem + OFF_B1;
    float* sW2  = smem + OFF_W2;
    float* sB2  = smem + OFF_B2;
    int*   sPIDX = (int*)(smem + OFF_PIDX);
    float* sYZ  = smem + OFF_YZ;
    float* sH0  = smem + OFF_H0;

    const int tid  = threadIdx.x;
    const int lane = tid & 31;
    const int wv   = tid >> 5;
    const int half = lane >> 4;     // 0: lanes 0-15, 1: lanes 16-31
    const int lm   = lane & 15;
    const int kl   = half * 2;      // K sub-offset for A/B fragments (f32 layout)

    // ---- cooperative weight staging (once per block; weights stay hot in LDS) ----
    // W1 [s][h_in][n_out] -> sW1T [s][n_out][h_in] with padded stride
    for (int i = tid; i < STATE * HID * HID; i += BLOCK) {
        int s = i >> 12;
        int h = (i >> 6) & 63;
        int n = i & 63;
        sW1T[(s * HID + n) * PAD + h] = W1[i];
    }
    for (int i = tid; i < SZ_W0; i += BLOCK) sW0[i] = W0[i];
    for (int i = tid; i < STATE * HID; i += BLOCK) {
        sB0[i] = b0[i]; sB1[i] = b1[i]; sW2[i] = W2[i];
    }
    if (tid < STATE)     sB2[tid]   = b2[tid];
    if (tid < STATE * 3) sPIDX[tid] = pidx_g[tid];
    __syncthreads();

    // ---- stage this wave's 16 rows of y and z into LDS (contiguous 96 floats each) ----
    const int tileBase = ((int)blockIdx.x * WAVES + wv) * ROWS_PER_WAVE;
    float* yb = sYZ + wv * 192;                 // [0..95] = y rows, [96..191] = z rows
    for (int i = lane; i < 96; i += 32) {
        int r = i / 6;
        int c = i - r * 6;
        int gr = tileBase + r; if (gr >= Bn) gr = Bn - 1;
        yb[i]      = y[(size_t)gr * 6 + c];
        yb[96 + i] = z[(size_t)gr * 6 + c];
    }
    float* h0w = sH0 + wv * (ROWS_PER_WAVE * PAD);

#pragma unroll 1
    for (int s = 0; s < STATE; ++s) {
        const int p0 = sPIDX[s * 3 + 0];
        const int p1 = sPIDX[s * 3 + 1];
        const int p2 = sPIDX[s * 3 + 2];

        // ---- layer 0: A = x tile (16x4). A layout: lanes0-15 hold K0,K1; lanes16-31 K2,K3.
        float av0 = half ? yb[lm * 6 + p2]      : yb[lm * 6 + p0];
        float av1 = half ? yb[96 + lm * 6 + s]  : yb[lm * 6 + p1];
        v2f a0; a0.x = av0; a0.y = av1;

#pragma unroll
        for (int nt = 0; nt < 4; ++nt) {
            const int n = nt * 16 + lm;
            v2f bf;                          // B 4x16: lane holds col n, rows kl, kl+1
            bf.x = sW0[(s * 4 + kl)     * HID + n];
            bf.y = sW0[(s * 4 + kl + 1) * HID + n];
            // C pre-loaded with bias -> WMMA performs x*W0 + b0 in one op
            const float bias = sB0[s * HID + n];
            v8f acc;
#pragma unroll
            for (int r = 0; r < 8; ++r) acc[r] = bias;
            acc = __builtin_amdgcn_wmma_f32_16x16x4_f32(
                false, a0, false, bf, (short)0, acc, false, false);
#pragma unroll
            for (int r = 0; r < 8; ++r)
                h0w[(r + half * 8) * PAD + n] = leaky(acc[r]);   // C layout -> LDS
        }

        // ---- layer 1 (64x64): kc-outer so each A fragment feeds 4 WMMAs ----
        v8f acc1[4];
#pragma unroll
        for (int nt = 0; nt < 4; ++nt) {
            const float bias = sB1[s * HID + nt * 16 + lm];
#pragma unroll
            for (int r = 0; r < 8; ++r) acc1[nt][r] = bias;      // C = b1
        }
        const float* arow = h0w + lm * PAD + kl;                 // A frag base: row lm
#pragma unroll
        for (int kc = 0; kc < 16; ++kc) {
            v2f af = *(const v2f*)(arow + kc * 4);               // one ds_load_b64
#pragma unroll
            for (int nt = 0; nt < 4; ++nt) {
                const float* brow = sW1T + (s * HID + nt * 16 + lm) * PAD + kl;
                v2f bf = *(const v2f*)(brow + kc * 4);           // ds_load_b64
                acc1[nt] = __builtin_amdgcn_wmma_f32_16x16x4_f32(
                    false, af, false, bf, (short)0, acc1[nt], false, false);
            }
        }

        // ---- layer 2 partials: leaky(h1) dotted with w2 along lane-resident n ----
        float pr[8] = {0.f, 0.f, 0.f, 0.f, 0.f, 0.f, 0.f, 0.f};
#pragma unroll
        for (int nt = 0; nt < 4; ++nt) {
            const float w2v = sW2[s * HID + nt * 16 + lm];
#pragma unroll
            for (int r = 0; r < 8; ++r)
                pr[r] += leaky(acc1[nt][r]) * w2v;
        }

        // ---- reduce N dimension (16 lanes per half) via xor butterfly ----
#pragma unroll
        for (int r = 0; r < 8; ++r) {
#pragma unroll
            for (int m = 1; m < 16; m <<= 1)
                pr[r] += __shfl_xor(pr[r], m, 32);
        }
        // lane lm (<8) in each half stores row (half*8 + lm)
        float val = pr[0];
#pragma unroll
        for (int r = 1; r < 8; ++r) val = (lm == r) ? pr[r] : val;
        const int gRow = tileBase + half * 8 + lm;
        if (lm < 8 && gRow < Bn)
            out[(size_t)gRow * 6 + s] = val + sB2[s];
    }
}

extern "C" void kernel_launch(void* const* d_in, const int* in_sizes, int n_in,
                              void* d_out, int out_size, void* d_ws, size_t ws_size,
                              hipStream_t stream) {
    (void)n_in; (void)out_size; (void)d_ws; (void)ws_size;
    const float* y    = (const float*)d_in[0];
    const float* z    = (const float*)d_in[1];
    // d_in[2] = index (unused by the reference output)
    const int*   pidx = (const int*)d_in[3];
    const float* W0   = (const float*)d_in[4];
    const float* b0   = (const float*)d_in[5];
    const float* W1   = (const float*)d_in[6];
    const float* b1   = (const float*)d_in[7];
    const float* W2   = (const float*)d_in[8];
    const float* b2   = (const float*)d_in[9];
    float* out = (float*)d_out;

    const int Bn = in_sizes[0] / STATE;                      // 1,000,000
    const int blocks = (Bn + ROWS_PER_BLOCK - 1) / ROWS_PER_BLOCK;  // 15625
    const size_t shmem = (size_t)SMEM_FLOATS * sizeof(float);       // ~129 KB

    scm_mlp_wmma_kernel<<<blocks, BLOCK, shmem, stream>>>(
        y, z, pidx, W0, b0, W1, b1, W2, b2, out, Bn);
}